// VariationalGCNEncoder_38929583571365
// MI455X (gfx1250) — compile-verified
//
#include <hip/hip_runtime.h>
#include <cstddef>

#define N_NODES 50000
#define N_EDGES 800000
#define IN_CH 128
#define HID 128
#define OUT_CH 64

typedef float v2f __attribute__((ext_vector_type(2)));
typedef float v8f __attribute__((ext_vector_type(8)));

// ---------------------------------------------------------------------------
// 1) Degree count (800k int atomics -- cheap, replaces 102M float atomics)
// ---------------------------------------------------------------------------
__global__ void count_deg_kernel(const long long* __restrict__ ei, int* __restrict__ deg) {
  int e = blockIdx.x * blockDim.x + threadIdx.x;
  if (e < N_EDGES) atomicAdd(&deg[(int)ei[N_EDGES + e]], 1);
}

// ---------------------------------------------------------------------------
// 2) Exclusive prefix sum over degrees -> CSR row offsets (+cursor copy),
//    also emits 1/max(deg,1) for the mean.
// ---------------------------------------------------------------------------
__global__ void scan_kernel(const int* __restrict__ deg, int* __restrict__ offs,
                            int* __restrict__ cursor, float* __restrict__ invd) {
  __shared__ int tmp[1024];
  __shared__ int carry;
  if (threadIdx.x == 0) carry = 0;
  __syncthreads();
  for (int base = 0; base < N_NODES; base += 1024) {
    int i = base + (int)threadIdx.x;
    int v = (i < N_NODES) ? deg[i] : 0;
    tmp[threadIdx.x] = v;
    __syncthreads();
    for (int off = 1; off < 1024; off <<= 1) {
      int t = (threadIdx.x >= (unsigned)off) ? tmp[threadIdx.x - off] : 0;
      __syncthreads();
      tmp[threadIdx.x] += t;
      __syncthreads();
    }
    if (i < N_NODES) {
      int o = carry + tmp[threadIdx.x] - v;   // exclusive
      offs[i] = o;
      cursor[i] = o;
      invd[i] = 1.0f / (float)(v > 1 ? v : 1);
    }
    __syncthreads();
    if (threadIdx.x == 1023) carry += tmp[1023];
    __syncthreads();
  }
  if (threadIdx.x == 0) offs[N_NODES] = carry;
}

// ---------------------------------------------------------------------------
// 3) Scatter edges into CSR-by-destination
// ---------------------------------------------------------------------------
__global__ void fill_csr_kernel(const long long* __restrict__ ei, int* __restrict__ cursor,
                                int* __restrict__ csr) {
  int e = blockIdx.x * blockDim.x + threadIdx.x;
  if (e < N_EDGES) {
    int d = (int)ei[N_EDGES + e];
    int slot = atomicAdd(&cursor[d], 1);
    csr[slot] = (int)ei[e];
  }
}

// ---------------------------------------------------------------------------
// 4) Neighbor sum: one wave32 per destination node, lane owns 4 channels.
//    Coalesced 512B/edge reads; sources are L2-resident (x/h = 25.6MB < 192MB).
//    No feature atomics at all.
// ---------------------------------------------------------------------------
__global__ void aggregate_kernel(const float* __restrict__ feat, const int* __restrict__ offs,
                                 const int* __restrict__ csr, float* __restrict__ agg) {
  int node = (int)((blockIdx.x * blockDim.x + threadIdx.x) >> 5);   // wave32
  int lane = threadIdx.x & 31;
  if (node >= N_NODES) return;
  int beg = offs[node], end = offs[node + 1];
  float4 acc = make_float4(0.f, 0.f, 0.f, 0.f);
  for (int i = beg; i < end; ++i) {
    int s = csr[i];
    float4 v = *(const float4*)(feat + (size_t)s * IN_CH + lane * 4);
    acc.x += v.x; acc.y += v.y; acc.z += v.z; acc.w += v.w;
  }
  *(float4*)(agg + (size_t)node * IN_CH + lane * 4) = acc;
}

// ---------------------------------------------------------------------------
// 5) Layer-1 fused GEMM: h = relu( (agg/deg) @ Wl + x @ Wr + b ), exact fp32
//    via V_WMMA_F32_16X16X4_F32. One wave per 16x16 tile; 8 waves = 128 cols.
// ---------------------------------------------------------------------------
__global__ void sage_gemm_relu_kernel(const float* __restrict__ agg, const float* __restrict__ selfx,
                                      const float* __restrict__ invd,
                                      const float* __restrict__ wl, const float* __restrict__ wr,
                                      const float* __restrict__ bias, float* __restrict__ outh) {
  const int lane = (int)(threadIdx.x & 31);
  const int wv   = (int)(threadIdx.x >> 5);        // column tile 0..7
  const int r0   = (int)blockIdx.x * 16;           // 3125 * 16 == 50000 exactly
  const int m    = lane & 15;
  const int hi   = lane >> 4;
  const int row  = r0 + m;
  const int col  = wv * 16 + m;
  const float inv = invd[row];
  const float* aRow = agg   + (size_t)row * IN_CH;
  const float* xRow = selfx + (size_t)row * IN_CH;
  v8f acc = {};
#pragma unroll 8
  for (int k0 = 0; k0 < IN_CH; k0 += 4) {
    const int ka = k0 + 2 * hi;
    v2f a = *(const v2f*)(aRow + ka);
    a.x *= inv; a.y *= inv;
    v2f b;
    b.x = wl[(size_t)ka * HID + col];
    b.y = wl[(size_t)(ka + 1) * HID + col];
    acc = __builtin_amdgcn_wmma_f32_16x16x4_f32(false, a, false, b, (short)0, acc, false, false);
  }
#pragma unroll 8
  for (int k0 = 0; k0 < IN_CH; k0 += 4) {
    const int ka = k0 + 2 * hi;
    v2f a = *(const v2f*)(xRow + ka);
    v2f b;
    b.x = wr[(size_t)ka * HID + col];
    b.y = wr[(size_t)(ka + 1) * HID + col];
    acc = __builtin_amdgcn_wmma_f32_16x16x4_f32(false, a, false, b, (short)0, acc, false, false);
  }
  const float bs = bias[col];
#pragma unroll
  for (int r = 0; r < 8; ++r) {
    float v = acc[r] + bs;
    v = v > 0.f ? v : 0.f;
    outh[(size_t)(r0 + r + 8 * hi) * HID + col] = v;
  }
}

// ---------------------------------------------------------------------------
// 6) Heads: mu / logstd share agg(h); blockIdx.y selects the weight set.
//    out = (agg/deg) @ Wl + h @ Wr + b  (no relu), 4 waves = 64 cols.
// ---------------------------------------------------------------------------
__global__ void head_gemm_kernel(const float* __restrict__ agg, const float* __restrict__ hfeat,
                                 const float* __restrict__ invd,
                                 const float* __restrict__ wmu_l, const float* __restrict__ wmu_r,
                                 const float* __restrict__ bmu,
                                 const float* __restrict__ wls_l, const float* __restrict__ wls_r,
                                 const float* __restrict__ bls,
                                 float* __restrict__ out) {
  const bool is_ls = (blockIdx.y != 0);
  const float* wl   = is_ls ? wls_l : wmu_l;
  const float* wr   = is_ls ? wls_r : wmu_r;
  const float* bias = is_ls ? bls   : bmu;
  float* o = out + (is_ls ? (size_t)N_NODES * OUT_CH : (size_t)0);

  const int lane = (int)(threadIdx.x & 31);
  const int wv   = (int)(threadIdx.x >> 5);        // column tile 0..3
  const int r0   = (int)blockIdx.x * 16;
  const int m    = lane & 15;
  const int hi   = lane >> 4;
  const int row  = r0 + m;
  const int col  = wv * 16 + m;
  const float inv = invd[row];
  const float* aRow = agg   + (size_t)row * HID;
  const float* xRow = hfeat + (size_t)row * HID;
  v8f acc = {};
#pragma unroll 8
  for (int k0 = 0; k0 < HID; k0 += 4) {
    const int ka = k0 + 2 * hi;
    v2f a = *(const v2f*)(aRow + ka);
    a.x *= inv; a.y *= inv;
    v2f b;
    b.x = wl[(size_t)ka * OUT_CH + col];
    b.y = wl[(size_t)(ka + 1) * OUT_CH + col];
    acc = __builtin_amdgcn_wmma_f32_16x16x4_f32(false, a, false, b, (short)0, acc, false, false);
  }
#pragma unroll 8
  for (int k0 = 0; k0 < HID; k0 += 4) {
    const int ka = k0 + 2 * hi;
    v2f a = *(const v2f*)(xRow + ka);
    v2f b;
    b.x = wr[(size_t)ka * OUT_CH + col];
    b.y = wr[(size_t)(ka + 1) * OUT_CH + col];
    acc = __builtin_amdgcn_wmma_f32_16x16x4_f32(false, a, false, b, (short)0, acc, false, false);
  }
  const float bs = bias[col];
#pragma unroll
  for (int r = 0; r < 8; ++r) {
    o[(size_t)(r0 + r + 8 * hi) * OUT_CH + col] = acc[r] + bs;
  }
}

// ---------------------------------------------------------------------------
extern "C" void kernel_launch(void* const* d_in, const int* in_sizes, int n_in,
                              void* d_out, int out_size, void* d_ws, size_t ws_size,
                              hipStream_t stream) {
  const float*     x    = (const float*)d_in[0];
  const long long* ei   = (const long long*)d_in[1];   // int64 edge_index [2, E]
  const float*     w1l  = (const float*)d_in[2];
  const float*     w1r  = (const float*)d_in[3];
  const float*     b1   = (const float*)d_in[4];
  const float*     wmul = (const float*)d_in[5];
  const float*     wmur = (const float*)d_in[6];
  const float*     bmu  = (const float*)d_in[7];
  const float*     wlsl = (const float*)d_in[8];
  const float*     wlsr = (const float*)d_in[9];
  const float*     bls  = (const float*)d_in[10];
  float* out = (float*)d_out;                           // [mu | logstd] flat

  // Workspace carve-up (~55 MB)
  float* h    = (float*)d_ws;                           // 50000*128 f32
  float* agg  = h    + (size_t)N_NODES * HID;           // 50000*128 f32
  float* invd = agg  + (size_t)N_NODES * HID;           // 50000 f32
  int*   deg    = (int*)(invd + N_NODES);               // 50000
  int*   offs   = deg    + N_NODES;                     // 50001
  int*   cursor = offs   + (N_NODES + 1);               // 50000
  int*   csr    = cursor + N_NODES;                     // 800000

  hipMemsetAsync(deg, 0, N_NODES * sizeof(int), stream);
  count_deg_kernel<<<(N_EDGES + 255) / 256, 256, 0, stream>>>(ei, deg);
  scan_kernel<<<1, 1024, 0, stream>>>(deg, offs, cursor, invd);
  fill_csr_kernel<<<(N_EDGES + 255) / 256, 256, 0, stream>>>(ei, cursor, csr);

  // agg(x) -> layer1 -> h
  aggregate_kernel<<<N_NODES / 8, 256, 0, stream>>>(x, offs, csr, agg);
  sage_gemm_relu_kernel<<<N_NODES / 16, 256, 0, stream>>>(agg, x, invd, w1l, w1r, b1, h);

  // agg(h) shared by both heads
  aggregate_kernel<<<N_NODES / 8, 256, 0, stream>>>(h, offs, csr, agg);
  dim3 g2(N_NODES / 16, 2);
  head_gemm_kernel<<<g2, 128, 0, stream>>>(agg, h, invd, wmul, wmur, bmu, wlsl, wlsr, bls, out);
}